// CrossModalAttention_37168646979931
// MI455X (gfx1250) — compile-verified
//
#include <hip/hip_runtime.h>

#define N_NODES 20000
#define N_EDGES 320000
#define SCALE_ATT 0.25f

typedef __attribute__((ext_vector_type(2))) float v2f;
typedef __attribute__((ext_vector_type(8))) float v8f;

// ---- monotonic float <-> ordered-uint mapping (for atomicMax on floats) ----
__device__ __forceinline__ unsigned f2o(float f) {
  unsigned u = __float_as_uint(f);
  return (u & 0x80000000u) ? ~u : (u | 0x80000000u);
}
__device__ __forceinline__ float o2f(unsigned o) {
  unsigned u = (o & 0x80000000u) ? (o & 0x7fffffffu) : ~o;
  return __uint_as_float(u);
}

// ---- CDNA5 async global->LDS copy (ASYNCcnt-tracked, no VGPR bounce) ----
__device__ __forceinline__ unsigned lds_off_of(void* p) {
  // __shared__ pointers round-trip generic -> addrspace(3); ptrtoint of the
  // AS3 pointer is the 32-bit LDS byte offset the async engine wants.
  return (unsigned)(uintptr_t)(__attribute__((address_space(3))) void*)p;
}
__device__ __forceinline__ void async_g2l_b128(unsigned lds_off, const void* gsrc) {
  asm volatile("global_load_async_to_lds_b128 %0, %1, off"
               :: "v"(lds_off), "v"((unsigned long long)(uintptr_t)gsrc)
               : "memory");
}
__device__ __forceinline__ void wait_async0() {
  asm volatile("s_wait_asynccnt 0" ::: "memory");
}

// ---------------------------------------------------------------------------
// WMMA f32 16x16 output tile: one wave computes 16 rows x 16 cols, K-loop of 4.
// A (16xK) is staged in LDS row-major [16][K]; B = W row-major [K][ldw].
// A frag layout (ISA 7.12.2): lanes 0-15 -> M=lane, VGPR0=K0,VGPR1=K1;
//                              lanes 16-31 -> M=lane-16, VGPR0=K2,VGPR1=K3.
// C/D layout: VGPR t = row t (lanes 0-15) / row t+8 (lanes 16-31).
// ---------------------------------------------------------------------------
__device__ __forceinline__ void wmma_tile(
    const float* __restrict__ ldsA, int K,
    const float* __restrict__ W, int ldw,
    const float* __restrict__ bias,
    float* __restrict__ out, int ldo,
    const float* __restrict__ resid,
    int row0, int c0, int lane)
{
  v8f acc = {};
  const int r  = lane & 15;
  const int hi = lane >> 4;
  for (int k = 0; k < K; k += 4) {
    const int ka = k + hi * 2;
    v2f a, b;
    a[0] = ldsA[r * K + ka];
    a[1] = ldsA[r * K + ka + 1];
    b[0] = W[ka * ldw + c0 + r];
    b[1] = W[(ka + 1) * ldw + c0 + r];
    acc = __builtin_amdgcn_wmma_f32_16x16x4_f32(false, a, false, b,
                                                (short)0, acc, false, false);
  }
  const float bv = bias[c0 + r];
#pragma unroll
  for (int t = 0; t < 8; ++t) {
    const int rr  = t + hi * 8;
    const int idx = (row0 + rr) * ldo + c0 + r;
    float v = acc[t] + bv;
    if (resid) v += resid[idx];
    out[idx] = v;
  }
}

// ---- zero-init of softmax/aggregation accumulators ----
__global__ void init_kernel(float* __restrict__ p, int n) {
  int i = blockIdx.x * blockDim.x + threadIdx.x;
  if (i < n) p[i] = 0.0f;
}

// ---- node projections: q/k/v = x @ W + b  (x: [N,128], W: [128,64]) ----
__global__ __launch_bounds__(128) void node_gemm_kernel(
    const float* __restrict__ x,
    const float* __restrict__ Wq, const float* __restrict__ bq,
    const float* __restrict__ Wk, const float* __restrict__ bk,
    const float* __restrict__ Wv, const float* __restrict__ bv,
    float* __restrict__ oq, float* __restrict__ ok, float* __restrict__ ov)
{
  __shared__ float tile[16 * 128];
  const int row0 = blockIdx.x * 16;
  __builtin_prefetch(Wq, 0, 1);
  __builtin_prefetch(Wk, 0, 1);
  __builtin_prefetch(Wv, 0, 1);
  // async stage A tile (8 KB): 512 x b128, 4 per thread
  {
    const float* src = x + (size_t)row0 * 128;
    const unsigned lbase = lds_off_of(tile);
    for (int t = threadIdx.x; t < 16 * 128 / 4; t += 128)
      async_g2l_b128(lbase + t * 16u, src + t * 4);
    wait_async0();
  }
  __syncthreads();
  const int lane = threadIdx.x & 31;
  const int c0   = (threadIdx.x >> 5) * 16;   // 4 waves -> 4 col tiles of 64
  wmma_tile(tile, 128, Wq, 64, bq, oq, 64, nullptr, row0, c0, lane);
  wmma_tile(tile, 128, Wk, 64, bk, ok, 64, nullptr, row0, c0, lane);
  wmma_tile(tile, 128, Wv, 64, bv, ov, 64, nullptr, row0, c0, lane);
}

// ---- edge projections: q/k/v = edge_attr @ W + b  (ea: [E,64], W: [64,64]) ----
__global__ __launch_bounds__(128) void edge_gemm_kernel(
    const float* __restrict__ ea,
    const float* __restrict__ Wq, const float* __restrict__ bq,
    const float* __restrict__ Wk, const float* __restrict__ bk,
    const float* __restrict__ Wv, const float* __restrict__ bv,
    float* __restrict__ oq, float* __restrict__ ok, float* __restrict__ ov)
{
  __shared__ float tile[16 * 64];
  const int row0 = blockIdx.x * 16;
  __builtin_prefetch(Wq, 0, 1);
  __builtin_prefetch(Wk, 0, 1);
  __builtin_prefetch(Wv, 0, 1);
  // async stage A tile (4 KB): 256 x b128, 2 per thread
  {
    const float* src = ea + (size_t)row0 * 64;
    const unsigned lbase = lds_off_of(tile);
    for (int t = threadIdx.x; t < 16 * 64 / 4; t += 128)
      async_g2l_b128(lbase + t * 16u, src + t * 4);
    wait_async0();
  }
  __syncthreads();
  const int lane = threadIdx.x & 31;
  const int c0   = (threadIdx.x >> 5) * 16;
  wmma_tile(tile, 64, Wq, 64, bq, oq, 64, nullptr, row0, c0, lane);
  wmma_tile(tile, 64, Wk, 64, bk, ok, 64, nullptr, row0, c0, lane);
  wmma_tile(tile, 64, Wv, 64, bv, ov, 64, nullptr, row0, c0, lane);
}

// ---- phase 1: endpoint scores s1[i,h] and global per-head max ----
__global__ __launch_bounds__(256) void p1_scores_kernel(
    const float* __restrict__ edge_q, const float* __restrict__ node_k,
    const int* __restrict__ row, const int* __restrict__ col,
    float* __restrict__ s1, unsigned* __restrict__ hmax)
{
  __shared__ unsigned lmax[4];
  if (threadIdx.x < 4) lmax[threadIdx.x] = 0u;
  __syncthreads();
  const int i = blockIdx.x * blockDim.x + threadIdx.x;
  if (i < 2 * N_EDGES) {
    const int e = (i < N_EDGES) ? i : i - N_EDGES;
    const int u = (i < N_EDGES) ? row[e] : col[e];
    const float4* q = (const float4*)(edge_q + (size_t)e * 64);
    const float4* k = (const float4*)(node_k + (size_t)u * 64);
#pragma unroll
    for (int h = 0; h < 4; ++h) {
      float acc = 0.0f;
#pragma unroll
      for (int d = 0; d < 4; ++d) {
        float4 a = q[h * 4 + d], b = k[h * 4 + d];
        acc += a.x * b.x + a.y * b.y + a.z * b.z + a.w * b.w;
      }
      const float s = acc * SCALE_ATT;
      s1[(size_t)i * 4 + h] = s;
      atomicMax(&lmax[h], f2o(s));
    }
  }
  __syncthreads();
  if (threadIdx.x < 4) atomicMax(&hmax[threadIdx.x], lmax[threadIdx.x]);
}

// ---- phase 1: exp / global sum / weighted-V pooling ----
__global__ __launch_bounds__(256) void p1_pool_kernel(
    const float* __restrict__ s1, const float* __restrict__ node_v,
    const int* __restrict__ row, const int* __restrict__ col,
    const unsigned* __restrict__ hmax,
    float* __restrict__ sum_p, float* __restrict__ numer)
{
  __shared__ float lsum[4];
  __shared__ float lnum[64];
  if (threadIdx.x < 4)  lsum[threadIdx.x] = 0.0f;
  if (threadIdx.x < 64) lnum[threadIdx.x] = 0.0f;
  __syncthreads();
  const int i = blockIdx.x * blockDim.x + threadIdx.x;
  if (i < 2 * N_EDGES) {
    const int e = (i < N_EDGES) ? i : i - N_EDGES;
    const int u = (i < N_EDGES) ? row[e] : col[e];
    const float* v = node_v + (size_t)u * 64;
#pragma unroll
    for (int h = 0; h < 4; ++h) {
      const float p = expf(s1[(size_t)i * 4 + h] - o2f(hmax[h]));
      atomicAdd(&lsum[h], p);
#pragma unroll
      for (int d = 0; d < 16; ++d)
        atomicAdd(&lnum[h * 16 + d], p * v[h * 16 + d]);
    }
  }
  __syncthreads();
  if (threadIdx.x < 4)  atomicAdd(&sum_p[threadIdx.x], lsum[threadIdx.x]);
  if (threadIdx.x < 64) atomicAdd(&numer[threadIdx.x], lnum[threadIdx.x]);
}

// ---- phase 1: outvec = (numer/sum_p) @ Woe + boe  (single 64-thread block) ----
__global__ __launch_bounds__(64) void p1_outvec_kernel(
    const float* __restrict__ numer, const float* __restrict__ sum_p,
    const float* __restrict__ Woe, const float* __restrict__ boe,
    float* __restrict__ outvec)
{
  __shared__ float pooled[64];
  const int t = threadIdx.x;
  pooled[t] = numer[t] / sum_p[t >> 4];
  __syncthreads();
  float acc = boe[t];
  for (int j = 0; j < 64; ++j) acc += pooled[j] * Woe[j * 64 + t];
  outvec[t] = acc;
}

// ---- phase 1: updated_edge = edge_attr + outvec (broadcast) ----
__global__ __launch_bounds__(256) void p1_edge_out_kernel(
    const float* __restrict__ ea, const float* __restrict__ outvec,
    float* __restrict__ out_edge)
{
  __shared__ float4 ov[16];
  if (threadIdx.x < 16) ov[threadIdx.x] = ((const float4*)outvec)[threadIdx.x];
  __syncthreads();
  const int i = blockIdx.x * blockDim.x + threadIdx.x;   // one float4 each
  if (i < N_EDGES * 16) {
    float4 a = ((const float4*)ea)[i];
    float4 b = ov[i & 15];
    float4 r; r.x = a.x + b.x; r.y = a.y + b.y; r.z = a.z + b.z; r.w = a.w + b.w;
    ((float4*)out_edge)[i] = r;
  }
}

// ---- phase 2: endpoint scores s2[i,h] and per-node per-head max ----
__global__ __launch_bounds__(256) void p2_scores_kernel(
    const float* __restrict__ node_q, const float* __restrict__ edge_k,
    const int* __restrict__ row, const int* __restrict__ col,
    float* __restrict__ s2, unsigned* __restrict__ smax)
{
  const int i = blockIdx.x * blockDim.x + threadIdx.x;
  if (i >= 2 * N_EDGES) return;
  const int e = (i < N_EDGES) ? i : i - N_EDGES;
  const int n = (i < N_EDGES) ? row[e] : col[e];
  const float4* q = (const float4*)(node_q + (size_t)n * 64);
  const float4* k = (const float4*)(edge_k + (size_t)e * 64);
#pragma unroll
  for (int h = 0; h < 4; ++h) {
    float acc = 0.0f;
#pragma unroll
    for (int d = 0; d < 4; ++d) {
      float4 a = q[h * 4 + d], b = k[h * 4 + d];
      acc += a.x * b.x + a.y * b.y + a.z * b.z + a.w * b.w;
    }
    const float s = acc * SCALE_ATT;
    s2[(size_t)i * 4 + h] = s;
    atomicMax(&smax[(size_t)n * 4 + h], f2o(s));
  }
}

// ---- phase 2: e = exp(s - smax[seg]), denom segment-sum ----
__global__ __launch_bounds__(256) void p2_exp_kernel(
    float* __restrict__ s2, const unsigned* __restrict__ smax,
    const int* __restrict__ row, const int* __restrict__ col,
    float* __restrict__ denom)
{
  const int i = blockIdx.x * blockDim.x + threadIdx.x;
  if (i >= 2 * N_EDGES) return;
  const int e = (i < N_EDGES) ? i : i - N_EDGES;
  const int n = (i < N_EDGES) ? row[e] : col[e];
#pragma unroll
  for (int h = 0; h < 4; ++h) {
    const float p = expf(s2[(size_t)i * 4 + h] - o2f(smax[(size_t)n * 4 + h]));
    s2[(size_t)i * 4 + h] = p;
    atomicAdd(&denom[(size_t)n * 4 + h], p);
  }
}

// ---- phase 2: agg[n] += (p/denom) * edge_v  (one thread per endpoint*head) ----
__global__ __launch_bounds__(256) void p2_agg_kernel(
    const float* __restrict__ s2, const float* __restrict__ denom,
    const float* __restrict__ edge_v,
    const int* __restrict__ row, const int* __restrict__ col,
    float* __restrict__ agg)
{
  const int gid = blockIdx.x * blockDim.x + threadIdx.x;
  if (gid >= 2 * N_EDGES * 4) return;
  const int i = gid >> 2;
  const int h = gid & 3;
  const int e = (i < N_EDGES) ? i : i - N_EDGES;
  const int n = (i < N_EDGES) ? row[e] : col[e];
  const float w2 = s2[(size_t)i * 4 + h] / fmaxf(denom[(size_t)n * 4 + h], 1e-20f);
  const float* v = edge_v + (size_t)e * 64 + h * 16;
  float* a = agg + (size_t)n * 64 + h * 16;
#pragma unroll
  for (int d = 0; d < 16; ++d) atomicAdd(&a[d], w2 * v[d]);
}

// ---- output projection: updated_x = agg @ Won + bon + x ----
__global__ __launch_bounds__(256) void out_gemm_kernel(
    const float* __restrict__ agg, const float* __restrict__ Won,
    const float* __restrict__ bon, const float* __restrict__ x,
    float* __restrict__ out_x)
{
  __shared__ float tile[16 * 64];
  const int row0 = blockIdx.x * 16;
  __builtin_prefetch(Won, 0, 1);
  // async stage A tile (4 KB): 256 x b128, 1 per thread
  {
    const float* src = agg + (size_t)row0 * 64;
    const unsigned lbase = lds_off_of(tile);
    for (int t = threadIdx.x; t < 16 * 64 / 4; t += 256)
      async_g2l_b128(lbase + t * 16u, src + t * 4);
    wait_async0();
  }
  __syncthreads();
  const int lane = threadIdx.x & 31;
  const int c0   = (threadIdx.x >> 5) * 16;   // 8 waves -> 8 col tiles of 128
  wmma_tile(tile, 64, Won, 128, bon, out_x, 128, x, row0, c0, lane);
}

extern "C" void kernel_launch(void* const* d_in, const int* in_sizes, int n_in,
                              void* d_out, int out_size, void* d_ws, size_t ws_size,
                              hipStream_t stream) {
  const float* x   = (const float*)d_in[0];
  const float* ea  = (const float*)d_in[1];
  const float* Wqn = (const float*)d_in[2];  const float* bqn = (const float*)d_in[3];
  const float* Wkn = (const float*)d_in[4];  const float* bkn = (const float*)d_in[5];
  const float* Wvn = (const float*)d_in[6];  const float* bvn = (const float*)d_in[7];
  const float* Wqe = (const float*)d_in[8];  const float* bqe = (const float*)d_in[9];
  const float* Wke = (const float*)d_in[10]; const float* bke = (const float*)d_in[11];
  const float* Wve = (const float*)d_in[12]; const float* bve = (const float*)d_in[13];
  const float* Won = (const float*)d_in[14]; const float* bon = (const float*)d_in[15];
  const float* Woe = (const float*)d_in[16]; const float* boe = (const float*)d_in[17];
  const int*   ei  = (const int*)d_in[18];
  const int* row = ei;
  const int* col = ei + N_EDGES;

  float* out_x = (float*)d_out;
  float* out_e = (float*)d_out + (size_t)N_NODES * 128;

  // ---- workspace layout (floats) ----
  float* ws = (float*)d_ws;
  size_t off = 0;
  float* node_q = ws + off; off += (size_t)N_NODES * 64;
  float* node_k = ws + off; off += (size_t)N_NODES * 64;
  float* node_v = ws + off; off += (size_t)N_NODES * 64;
  float* edge_q = ws + off; off += (size_t)N_EDGES * 64;
  float* edge_k = ws + off; off += (size_t)N_EDGES * 64;
  float* edge_v = ws + off; off += (size_t)N_EDGES * 64;
  float* s1     = ws + off; off += (size_t)2 * N_EDGES * 4;
  float* s2     = ws + off; off += (size_t)2 * N_EDGES * 4;
  size_t zero_begin = off;
  unsigned* smax = (unsigned*)(ws + off); off += (size_t)N_NODES * 4;
  float* denom  = ws + off; off += (size_t)N_NODES * 4;
  float* agg    = ws + off; off += (size_t)N_NODES * 64;
  unsigned* hmax = (unsigned*)(ws + off); off += 4;
  float* sum_p  = ws + off; off += 4;
  float* numer  = ws + off; off += 64;
  float* outvec = ws + off; off += 64;
  const int zlen = (int)(off - zero_begin);

  // 1) zero accumulators (smax/denom/agg/hmax/sum_p/numer/outvec)
  init_kernel<<<(zlen + 255) / 256, 256, 0, stream>>>(ws + zero_begin, zlen);

  // 2-3) projections via f32 WMMA (A tiles staged with async global->LDS)
  node_gemm_kernel<<<N_NODES / 16, 128, 0, stream>>>(
      x, Wqn, bqn, Wkn, bkn, Wvn, bvn, node_q, node_k, node_v);
  edge_gemm_kernel<<<N_EDGES / 16, 128, 0, stream>>>(
      ea, Wqe, bqe, Wke, bke, Wve, bve, edge_q, edge_k, edge_v);

  // 4-7) phase 1: global softmax over 2E endpoints, pooled vector, edge update
  const int ep_blocks = (2 * N_EDGES + 255) / 256;
  p1_scores_kernel<<<ep_blocks, 256, 0, stream>>>(edge_q, node_k, row, col, s1, hmax);
  p1_pool_kernel<<<ep_blocks, 256, 0, stream>>>(s1, node_v, row, col, hmax, sum_p, numer);
  p1_outvec_kernel<<<1, 64, 0, stream>>>(numer, sum_p, Woe, boe, outvec);
  p1_edge_out_kernel<<<(N_EDGES * 16 + 255) / 256, 256, 0, stream>>>(ea, outvec, out_e);

  // 8-10) phase 2: per-node segment softmax and aggregation
  p2_scores_kernel<<<ep_blocks, 256, 0, stream>>>(node_q, edge_k, row, col, s2, smax);
  p2_exp_kernel<<<ep_blocks, 256, 0, stream>>>(s2, smax, row, col, denom);
  p2_agg_kernel<<<(2 * N_EDGES * 4 + 255) / 256, 256, 0, stream>>>(
      s2, denom, edge_v, row, col, agg);

  // 11) output projection + residual via f32 WMMA
  out_gemm_kernel<<<N_NODES / 16, 256, 0, stream>>>(agg, Won, bon, x, out_x);
}